// SparseMaxPoolTestTorch_26963804684447
// MI455X (gfx1250) — compile-verified
//
#include <hip/hip_runtime.h>
#include <cstdint>

// out shape: (B=4, 32, 32, 32, C=64) f32, flat index = cell*64 + c,
// cell = ((b*32 + z>>2)*32 + y>>2)*32 + x>>2
static constexpr unsigned kNegInfBits = 0xFF800000u;  // -inf f32

// clang ext-vector types so __builtin_nontemporal_load emits b128 loads with th:TH_NT
typedef float v4f __attribute__((ext_vector_type(4)));
typedef int   v4i __attribute__((ext_vector_type(4)));
typedef unsigned v4u __attribute__((ext_vector_type(4)));

// ---------------------------------------------------------------- init: fill with -inf
__global__ void spool_init(v4u* __restrict__ out4, int n4) {
  int t = blockIdx.x * blockDim.x + threadIdx.x;
  if (t < n4) {
    v4u v = {kNegInfBits, kNegInfBits, kNegInfBits, kNegInfBits};
    out4[t] = v;  // regular-temporal: this region is the atomic working set, keep in L2
  }
}

// ------------------------------------------------- scatter-max via CDNA5 f32 max atomic
// thread t -> point n = t>>4, channel quad q = t&15 (4 channels per thread, b128 load).
// Lanes 0..15 of a wave cover one point's 64 channels contiguously (256B): feature loads
// and atomics are fully coalesced, no intra-wave address collisions.
// Streamed inputs are loaded non-temporally so the one-shot 136MB stream doesn't evict
// the 33.5MB L2-resident accumulator that the 32M atomics hammer.
__global__ void spool_scatter(const v4f* __restrict__ feat4,
                              const v4i* __restrict__ coors,
                              float* __restrict__ out, int total) {
  int t = blockIdx.x * blockDim.x + threadIdx.x;
  if (t < total) {
    int n = t >> 4;
    int q = t & 15;
    v4i co = __builtin_nontemporal_load(coors + n);  // b,z,y,x (NT b128, L0 broadcast)
    unsigned cell = (((unsigned)co.x * 32u + ((unsigned)co.y >> 2)) * 32u +
                     ((unsigned)co.z >> 2)) * 32u + ((unsigned)co.w >> 2);
    unsigned off = cell * 256u + (unsigned)q * 16u;  // byte offset into out
    v4f f = __builtin_nontemporal_load(feat4 + t);   // NT b128 feature stream
    // GVS mode: SGPR-pair base + 32-bit VGPR offset + literal offset.
    // No-return form -> STOREcnt only. DEV scope -> RMW in the device-coherent L2.
    asm volatile(
        "global_atomic_max_num_f32 %0, %2, %1 offset:0  scope:SCOPE_DEV\n\t"
        "global_atomic_max_num_f32 %0, %3, %1 offset:4  scope:SCOPE_DEV\n\t"
        "global_atomic_max_num_f32 %0, %4, %1 offset:8  scope:SCOPE_DEV\n\t"
        "global_atomic_max_num_f32 %0, %5, %1 offset:12 scope:SCOPE_DEV"
        :
        : "v"(off), "s"(out), "v"(f.x), "v"(f.y), "v"(f.z), "v"(f.w)
        : "memory");
  }
  // Compiler can't see the STOREcnt incremented by the asm atomics; make sure they
  // retire before s_endpgm / DEALLOC_VGPRS.
  asm volatile("s_wait_storecnt 0x0" ::: "memory");
}

// ---------------------------------------------- finalize: empty cells (-inf) become 0
__global__ void spool_finalize(v4u* __restrict__ out4, int n4) {
  int t = blockIdx.x * blockDim.x + threadIdx.x;
  if (t < n4) {
    v4u v = out4[t];  // L2-hot after the atomic phase
    if (v.x == kNegInfBits) v.x = 0u;
    if (v.y == kNegInfBits) v.y = 0u;
    if (v.z == kNegInfBits) v.z = 0u;
    if (v.w == kNegInfBits) v.w = 0u;
    out4[t] = v;
  }
}

extern "C" void kernel_launch(void* const* d_in, const int* in_sizes, int n_in,
                              void* d_out, int out_size, void* d_ws, size_t ws_size,
                              hipStream_t stream) {
  (void)d_ws; (void)ws_size; (void)n_in;
  const v4f* feat4 = (const v4f*)d_in[0];  // (N, 64) f32
  const v4i* coors = (const v4i*)d_in[1];  // (N, 4) i32: b,z,y,x
  float* out = (float*)d_out;              // (4,32,32,32,64) f32

  const int C = 64;
  const int N = in_sizes[0] / C;       // 500000
  const int n4 = out_size / 4;         // 2,097,152 x uint4
  const int threads = 256;             // 8 wave32s per block

  spool_init<<<(n4 + threads - 1) / threads, threads, 0, stream>>>((v4u*)out, n4);

  const int total = N * 16;            // one thread per (point, channel-quad)
  spool_scatter<<<(total + threads - 1) / threads, threads, 0, stream>>>(
      feat4, coors, out, total);

  spool_finalize<<<(n4 + threads - 1) / threads, threads, 0, stream>>>((v4u*)out, n4);
}